// FlashKAT_Group_25829933318783
// MI455X (gfx1250) — compile-verified
//
#include <hip/hip_runtime.h>
#include <stdint.h>

// Grouped rational (Pade) activation: y = P5(x) / (1 + |x * Q3(x)|), grouped coeffs.
// B=4, L=4096, D=2048, G=8. Pure streaming op: 268 MB traffic -> ~11.5 us floor
// at 23.3 TB/s HBM. WMMA inapplicable (no contraction; AI ~ 1.75 flop/byte).
// Strategy: gfx1250 async global->LDS B128 copies (ASYNCcnt), 4-deep LDS ring
// per wave, ds_load_b128 consume, v_fma Horner chain, v_rcp_f32, NT b128 stores.

typedef float v4f __attribute__((ext_vector_type(4)));

#define NBUF 4                      // async pipeline depth (LDS ring buffers)
#define TPB  512                    // threads per block == float4 per row (D/4)
#define D4   512                    // float4 per row (D = 2048)
#define DG4_SHIFT 6                 // 64 float4 per group (G = 8) -> tid >> 6
#define ROWBYTES ((uint64_t)D4 * 16)

// Issue one per-lane 16B async copy: global address gaddr to LDS offset lds_off. ASYNCcnt++.
__device__ __forceinline__ void async_copy_b128(uint32_t lds_off, uint64_t gaddr) {
    asm volatile("global_load_async_to_lds_b128 %0, %1, off"
                 :: "v"(lds_off), "v"(gaddr)
                 : "memory");
}

template <int N>
__device__ __forceinline__ void wait_async() {
    asm volatile("s_wait_asynccnt %0" :: "i"(N) : "memory");
}

__device__ __forceinline__ void wait_ds0() {
    asm volatile("s_wait_dscnt 0" ::: "memory");
}

__device__ __forceinline__ float rational1(float x,
        float a0, float a1, float a2, float a3, float a4, float a5,
        float b0, float b1, float b2, float b3) {
    float num = __builtin_fmaf(a5, x, a4);
    num = __builtin_fmaf(num, x, a3);
    num = __builtin_fmaf(num, x, a2);
    num = __builtin_fmaf(num, x, a1);
    num = __builtin_fmaf(num, x, a0);
    float z = __builtin_fmaf(b3, x, b2);
    z = __builtin_fmaf(z, x, b1);
    z = __builtin_fmaf(z, x, b0);
    float den = 1.0f + __builtin_fabsf(x * z);
    // den >= 1.0, so v_rcp_f32 is safe (~1 ulp); avoids the IEEE div sequence.
    return num * __builtin_amdgcn_rcpf(den);
}

__global__ __launch_bounds__(TPB) void kat_rational_kernel(
        const v4f* __restrict__ x,      // (rows, D4) float4
        const float* __restrict__ numv, // (G, 6)
        const float* __restrict__ denv, // (G, 4)
        v4f* __restrict__ y,            // (rows, D4)
        int rows, int rows_per_block) {
    __shared__ v4f buf[NBUF][TPB];     // 32 KB LDS ring

    const int tid = threadIdx.x;
    const int g = tid >> DG4_SHIFT;    // group fixed per thread for whole kernel

    // Coefficients live in registers; loads are broadcast (320 B total, L2-hot).
    const float a0 = numv[g * 6 + 0];
    const float a1 = numv[g * 6 + 1];
    const float a2 = numv[g * 6 + 2];
    const float a3 = numv[g * 6 + 3];
    const float a4 = numv[g * 6 + 4];
    const float a5 = numv[g * 6 + 5];
    const float b0 = denv[g * 4 + 0];
    const float b1 = denv[g * 4 + 1];
    const float b2 = denv[g * 4 + 2];
    const float b3 = denv[g * 4 + 3];

    int row0 = blockIdx.x * rows_per_block;
    int row_end = row0 + rows_per_block;
    if (row_end > rows) row_end = rows;
    if (row0 >= row_end) return;       // uniform per block
    const int n = row_end - row0;

    // Raw LDS byte offsets for this lane's slot in each ring buffer.
    uint32_t lds[NBUF];
#pragma unroll
    for (int k = 0; k < NBUF; ++k)
        lds[k] = (uint32_t)(uintptr_t)&buf[k][tid];

    auto consume = [&](int k, v4f* yptr) {
        v4f v = buf[k][tid];           // ds_load_b128
        v4f o;
        o.x = rational1(v.x, a0, a1, a2, a3, a4, a5, b0, b1, b2, b3);
        o.y = rational1(v.y, a0, a1, a2, a3, a4, a5, b0, b1, b2, b3);
        o.z = rational1(v.z, a0, a1, a2, a3, a4, a5, b0, b1, b2, b3);
        o.w = rational1(v.w, a0, a1, a2, a3, a4, a5, b0, b1, b2, b3);
        // Streaming output: NT hint so 268 MB stream doesn't thrash 192 MB L2.
        __builtin_nontemporal_store(o, yptr);
    };

    // Incremental pointers: no per-iteration multiplies.
    uint64_t gissue = (uint64_t)(uintptr_t)(x + (size_t)row0 * D4 + tid);
    v4f*     yptr   = y + (size_t)row0 * D4 + tid;

    // ---- Prologue: fill the pipeline (up to NBUF rows in flight per wave) ----
    const int npro = (n < NBUF) ? n : NBUF;
    for (int k = 0; k < npro; ++k)
        async_copy_b128(lds[k], gissue + (uint64_t)k * ROWBYTES);
    gissue += (uint64_t)NBUF * ROWBYTES;   // next row to issue

    if (n >= NBUF) {
        // ---- Steady state: wait oldest, consume, refill same buffer ----
        int k = 0;
        for (int cnt = n - NBUF; cnt > 0; --cnt) {
            wait_async<NBUF - 1>();    // oldest async copy (this buffer) done
            consume(k, yptr);
            wait_ds0();                // ds_load drained before async overwrite
            async_copy_b128(lds[k], gissue);
            gissue += ROWBYTES;
            yptr   += D4;
            k = (k + 1) & (NBUF - 1);
        }
        // ---- Epilogue: drain pipeline with decreasing split-waits ----
        wait_async<3>(); consume(k, yptr);                 yptr += D4; k = (k + 1) & 3;
        wait_async<2>(); consume(k, yptr);                 yptr += D4; k = (k + 1) & 3;
        wait_async<1>(); consume(k, yptr);                 yptr += D4; k = (k + 1) & 3;
        wait_async<0>(); consume(k, yptr);
    } else {
        // Tail-safe fallback (not hit with the launch config below).
        for (int i = 0; i < n; ++i) {
            wait_async<0>();
            consume(i & (NBUF - 1), yptr);
            yptr += D4;
        }
    }
}

extern "C" void kernel_launch(void* const* d_in, const int* in_sizes, int n_in,
                              void* d_out, int out_size, void* d_ws, size_t ws_size,
                              hipStream_t stream) {
    const v4f*   x    = (const v4f*)d_in[0];    // (B,L,D) fp32
    const float* numv = (const float*)d_in[1];  // (G,6)
    const float* denv = (const float*)d_in[2];  // (G,4)
    v4f*         y    = (v4f*)d_out;

    const int D    = 2048;
    const int rows = in_sizes[0] / D;     // B*L = 16384

    int blocks = 1024;
    if (blocks > rows) blocks = rows;
    const int rpb = (rows + blocks - 1) / blocks;   // 16 rows per block

    kat_rational_kernel<<<dim3(blocks), dim3(TPB), 0, stream>>>(
        x, numv, denv, y, rows, rpb);
}